// LGCN_mlp_18433999635010
// MI455X (gfx1250) — compile-verified
//
#include <hip/hip_runtime.h>
#include <hip/hip_bf16.h>

// ---------------------------------------------------------------------------
// LGCN (K=8 hops) + MLP for MI455X (gfx1250, wave32).
//  - Propagation: wave-per-edge scatter with hardware f32 global atomics
//    (coalesced 512B row segments: 32 lanes x 4 floats = 128 features).
//  - fc1 / fc2: f32 WMMA 16x16x4 (exact f32 math, matches reference),
//    LDS-staged tiles, fused bias + leaky-ReLU / BatchNorm.
//  - BatchNorm stats: deterministic two-stage block reduction.
// Workspace requirement: ~668 MB.
// ---------------------------------------------------------------------------

#define NN   100000     // nodes
#define EE   1600000    // edges
#define DD   128        // input feature width
#define KHOP 8
#define F1   1152       // DD*(KHOP+1)
#define HH   512        // hidden
#define OUTC 32         // output
#define SLOPE 0.2f
#define EPSB  1e-5f

typedef __attribute__((ext_vector_type(2))) float v2f;
typedef __attribute__((ext_vector_type(8))) float v8f;

// ---------------------------------------------------------------- degree ----
__global__ void k_deg_init(float* __restrict__ deg) {
    int i = blockIdx.x * 256 + threadIdx.x;
    if (i < NN) deg[i] = 1.0f;                       // self-loop
}

__global__ void k_deg_acc(const int* __restrict__ col, float* __restrict__ deg) {
    int e = blockIdx.x * 256 + threadIdx.x;
    if (e < EE) unsafeAtomicAdd(&deg[col[e]], 1.0f); // global_atomic_add_f32
}

__global__ void k_dinv(float* __restrict__ deg) {
    int i = blockIdx.x * 256 + threadIdx.x;
    if (i < NN) {
        float d = deg[i];
        deg[i] = (d > 0.f) ? rsqrtf(d) : 0.f;        // in-place -> dinv
    }
}

// ------------------------------------------------------------ propagation ---
// hop 0 slice of h <- feature   (N*32 float4 lanes)
__global__ void k_copy0(const float* __restrict__ feat, float* __restrict__ h) {
    int idx = blockIdx.x * 256 + threadIdx.x;        // NN*32 total
    int r = idx >> 5, j = (idx & 31) << 2;
    float4 v = *(const float4*)(feat + (size_t)r * DD + j);
    *(float4*)(h + (size_t)r * F1 + j) = v;
}

// slice hop+1 <- dinv[r]^2 * slice hop   (self-loop term, also zero-inits dst)
__global__ void k_selfloop(const float* __restrict__ dinv, float* __restrict__ h, int hop) {
    int idx = blockIdx.x * 256 + threadIdx.x;        // NN*32 total
    int r = idx >> 5, j = (idx & 31) << 2;
    float w = dinv[r] * dinv[r];
    float4 v = *(const float4*)(h + (size_t)r * F1 + (size_t)hop * DD + j);
    float4 o; o.x = w * v.x; o.y = w * v.y; o.z = w * v.z; o.w = w * v.w;
    *(float4*)(h + (size_t)r * F1 + (size_t)(hop + 1) * DD + j) = o;
}

// wave-per-edge: dst[col] += dinv[row]*dinv[col] * src[row]
__global__ void k_scatter(const int* __restrict__ rows, const int* __restrict__ cols,
                          const float* __restrict__ dinv, float* __restrict__ h, int hop) {
    int gid = blockIdx.x * 256 + threadIdx.x;
    int e = gid >> 5, lane = gid & 31;
    if (e >= EE) return;
    int r = rows[e], c = cols[e];
    float w = dinv[r] * dinv[c];
    const float* src = h + (size_t)r * F1 + (size_t)hop * DD;
    float* dst = h + (size_t)c * F1 + (size_t)(hop + 1) * DD;
#pragma unroll
    for (int j = 0; j < 4; ++j) {
        int f = lane + j * 32;
        unsafeAtomicAdd(&dst[f], w * src[f]);
    }
}

// ------------------------------------------------------------------- fc1 ----
// hmid[N,512] = leaky_relu(h[N,1152] @ W1 + b1)
// block = 256 thr (8 waves), tile 128 rows x 64 cols; wave w -> 16 rows x 64 cols.
__global__ __launch_bounds__(256) void k_fc1(const float* __restrict__ h,
                                             const float* __restrict__ W1,
                                             const float* __restrict__ b1,
                                             float* __restrict__ hmid) {
    __shared__ float As[128 * 33];   // stride 33: conflict-free frag reads
    __shared__ float Bs[32 * 80];    // stride 80: half-waves hit disjoint banks
    const int tid  = threadIdx.x;
    const int wave = tid >> 5, lane = tid & 31;
    const int half = lane >> 4, l16 = lane & 15;
    const int r0 = blockIdx.y * 128;
    const int n0 = blockIdx.x * 64;

    v8f acc[4];
#pragma unroll
    for (int t = 0; t < 4; ++t) acc[t] = (v8f){0, 0, 0, 0, 0, 0, 0, 0};

    for (int kb = 0; kb < F1; kb += 32) {
        __syncthreads();
        // A tile: 128x32 = 1024 float4, 4 per thread (coalesced)
#pragma unroll
        for (int i = 0; i < 4; ++i) {
            int idx = tid + i * 256;
            int row = idx >> 3, c4 = (idx & 7) << 2;
            int gr = r0 + row;
            float4 v = make_float4(0.f, 0.f, 0.f, 0.f);
            if (gr < NN) v = *(const float4*)(h + (size_t)gr * F1 + kb + c4);
            float* d = &As[row * 33 + c4];
            d[0] = v.x; d[1] = v.y; d[2] = v.z; d[3] = v.w;
        }
        // B tile: 32x64 = 512 float4, 2 per thread
#pragma unroll
        for (int i = 0; i < 2; ++i) {
            int idx = tid + i * 256;
            int row = idx >> 4, c4 = (idx & 15) << 2;
            float4 v = *(const float4*)(W1 + (size_t)(kb + row) * HH + n0 + c4);
            float* d = &Bs[row * 80 + c4];
            d[0] = v.x; d[1] = v.y; d[2] = v.z; d[3] = v.w;
        }
        __syncthreads();
#pragma unroll
        for (int kk = 0; kk < 32; kk += 4) {
            // A frag 16x4: lanes 0-15 K={kk,kk+1}, lanes 16-31 K={kk+2,kk+3}
            const float* ap = &As[(wave * 16 + l16) * 33 + kk + 2 * half];
            v2f a; a.x = ap[0]; a.y = ap[1];
#pragma unroll
            for (int t = 0; t < 4; ++t) {
                const float* bp = &Bs[(kk + 2 * half) * 80 + t * 16 + l16];
                v2f b; b.x = bp[0]; b.y = bp[80];
                acc[t] = __builtin_amdgcn_wmma_f32_16x16x4_f32(
                    false, a, false, b, (short)0, acc[t], false, false);
            }
        }
    }
    // epilogue: bias + leaky-ReLU; C layout: VGPR j -> M = j + 8*half, N = l16
#pragma unroll
    for (int t = 0; t < 4; ++t) {
        int c = n0 + t * 16 + l16;
        float bias = b1[c];
#pragma unroll
        for (int j = 0; j < 8; ++j) {
            int r = r0 + wave * 16 + half * 8 + j;
            if (r < NN) {
                float v = acc[t][j] + bias;
                v = (v >= 0.f) ? v : SLOPE * v;
                hmid[(size_t)r * HH + c] = v;
            }
        }
    }
}

// ------------------------------------------------------------- BN stats -----
// stage 1: 512 blocks, each writes per-channel partial sum/sumsq (deterministic)
__global__ __launch_bounds__(256) void k_stats_partial(const float* __restrict__ hmid,
                                                       float* __restrict__ part) {
    int c0 = threadIdx.x, c1 = threadIdx.x + 256;
    float s0 = 0.f, s1 = 0.f, q0 = 0.f, q1 = 0.f;
    for (int r = blockIdx.x; r < NN; r += gridDim.x) {
        const float* row = hmid + (size_t)r * HH;
        float a = row[c0], b = row[c1];
        s0 += a; q0 += a * a;
        s1 += b; q1 += b * b;
    }
    float* p = part + (size_t)blockIdx.x * 1024;
    p[c0] = s0; p[c1] = s1; p[512 + c0] = q0; p[512 + c1] = q1;
}

// stage 2: fold mean/var into per-channel scale/shift
__global__ void k_stats_final(const float* __restrict__ part,
                              const float* __restrict__ gamma,
                              const float* __restrict__ beta,
                              float* __restrict__ scale, float* __restrict__ shift) {
    int c = blockIdx.x * 256 + threadIdx.x;
    if (c >= HH) return;
    float s = 0.f, q = 0.f;
    for (int b = 0; b < 512; ++b) {
        s += part[(size_t)b * 1024 + c];
        q += part[(size_t)b * 1024 + 512 + c];
    }
    float mean = s / (float)NN;
    float var  = q / (float)NN - mean * mean;     // biased, matches jnp.var
    float inv  = rsqrtf(var + EPSB);
    float sc   = gamma[c] * inv;
    scale[c] = sc;
    shift[c] = beta[c] - mean * sc;
}

// ------------------------------------------------------------------- fc2 ----
// out[N,32] = (hmid*scale + shift) @ W2 + b2 ; BN applied while staging A.
__global__ __launch_bounds__(256) void k_fc2(const float* __restrict__ hmid,
                                             const float* __restrict__ scale,
                                             const float* __restrict__ shift,
                                             const float* __restrict__ W2,
                                             const float* __restrict__ b2,
                                             float* __restrict__ out) {
    __shared__ float As[128 * 33];
    __shared__ float Bs[32 * 40];    // stride 40: half-waves -> disjoint banks
    const int tid  = threadIdx.x;
    const int wave = tid >> 5, lane = tid & 31;
    const int half = lane >> 4, l16 = lane & 15;
    const int r0 = blockIdx.x * 128;

    v8f acc[2];
#pragma unroll
    for (int t = 0; t < 2; ++t) acc[t] = (v8f){0, 0, 0, 0, 0, 0, 0, 0};

    for (int kb = 0; kb < HH; kb += 32) {
        __syncthreads();
#pragma unroll
        for (int i = 0; i < 4; ++i) {
            int idx = tid + i * 256;
            int row = idx >> 3, c4 = (idx & 7) << 2;
            int gr = r0 + row;
            float4 v = make_float4(0.f, 0.f, 0.f, 0.f);
            if (gr < NN) v = *(const float4*)(hmid + (size_t)gr * HH + kb + c4);
            float4 sc = *(const float4*)(scale + kb + c4);
            float4 sh = *(const float4*)(shift + kb + c4);
            float* d = &As[row * 33 + c4];
            d[0] = v.x * sc.x + sh.x; d[1] = v.y * sc.y + sh.y;
            d[2] = v.z * sc.z + sh.z; d[3] = v.w * sc.w + sh.w;
        }
        {   // B tile 32x32 = 256 float4, 1 per thread
            int row = tid >> 3, c4 = (tid & 7) << 2;
            float4 v = *(const float4*)(W2 + (size_t)(kb + row) * OUTC + c4);
            float* d = &Bs[row * 40 + c4];
            d[0] = v.x; d[1] = v.y; d[2] = v.z; d[3] = v.w;
        }
        __syncthreads();
#pragma unroll
        for (int kk = 0; kk < 32; kk += 4) {
            const float* ap = &As[(wave * 16 + l16) * 33 + kk + 2 * half];
            v2f a; a.x = ap[0]; a.y = ap[1];
#pragma unroll
            for (int t = 0; t < 2; ++t) {
                const float* bp = &Bs[(kk + 2 * half) * 40 + t * 16 + l16];
                v2f b; b.x = bp[0]; b.y = bp[40];
                acc[t] = __builtin_amdgcn_wmma_f32_16x16x4_f32(
                    false, a, false, b, (short)0, acc[t], false, false);
            }
        }
    }
#pragma unroll
    for (int t = 0; t < 2; ++t) {
        int c = t * 16 + l16;
        float bias = b2[c];
#pragma unroll
        for (int j = 0; j < 8; ++j) {
            int r = r0 + wave * 16 + half * 8 + j;
            if (r < NN) out[(size_t)r * OUTC + c] = acc[t][j] + bias;
        }
    }
}

// ---------------------------------------------------------------- driver ----
extern "C" void kernel_launch(void* const* d_in, const int* in_sizes, int n_in,
                              void* d_out, int out_size, void* d_ws, size_t ws_size,
                              hipStream_t stream) {
    const float* feature = (const float*)d_in[0];
    const int*   edges   = (const int*)d_in[1];   // [0..E)=row(src), [E..2E)=col(dst)
    const float* W1      = (const float*)d_in[2];
    const float* b1      = (const float*)d_in[3];
    const float* gamma   = (const float*)d_in[4];
    const float* beta    = (const float*)d_in[5];
    const float* W2      = (const float*)d_in[6];
    const float* b2      = (const float*)d_in[7];
    float* out = (float*)d_out;

    // workspace layout (floats); total ~167M floats (~668 MB)
    float* ws = (float*)d_ws;
    size_t o = 0;
    float* dinv = ws + o; o += 100352;                 // deg -> dinv (in place)
    float* h    = ws + o; o += (size_t)NN * F1;        // [N,1152] concat hops
    float* hmid = ws + o; o += (size_t)NN * HH;        // [N,512] fc1 output
    float* part = ws + o; o += (size_t)512 * 1024;     // BN partials
    float* scale = ws + o; o += 512;
    float* shift = ws + o; o += 512;
    (void)ws_size; (void)n_in; (void)in_sizes; (void)out_size;

    const int*  ecol = edges + EE;
    const int*  erow = edges;

    k_deg_init<<<(NN + 255) / 256, 256, 0, stream>>>(dinv);
    k_deg_acc<<<(EE + 255) / 256, 256, 0, stream>>>(ecol, dinv);
    k_dinv<<<(NN + 255) / 256, 256, 0, stream>>>(dinv);

    k_copy0<<<NN * 32 / 256, 256, 0, stream>>>(feature, h);
    for (int hop = 0; hop < KHOP; ++hop) {
        k_selfloop<<<NN * 32 / 256, 256, 0, stream>>>(dinv, h, hop);
        k_scatter<<<EE * 32 / 256, 256, 0, stream>>>(erow, ecol, dinv, h, hop);
    }

    dim3 g1(HH / 64, (NN + 127) / 128);   // col-block fastest -> A reuse in L2
    k_fc1<<<g1, 256, 0, stream>>>(h, W1, b1, hmid);

    k_stats_partial<<<512, 256, 0, stream>>>(hmid, part);
    k_stats_final<<<2, 256, 0, stream>>>(part, gamma, beta, scale, shift);

    k_fc2<<<(NN + 127) / 128, 256, 0, stream>>>(hmid, scale, shift, W2, b2, out);
}